// EdgePropertyPredictionModel3_5360119185644
// MI455X (gfx1250) — compile-verified
//
#include <hip/hip_runtime.h>

typedef __attribute__((ext_vector_type(2))) float v2f;
typedef __attribute__((ext_vector_type(8))) float v8f;

#define GN 50000
#define GNE 800000

// ---------------------------------------------------------------------------
// Tiled f32 WMMA GEMM:  C[M,Nc] = affine(A[M,K]) @ W[K,Nc] + bias, opt. ReLU
// Block = 256 threads (8 waves). C tile = 128 rows x 16 cols, one 16x16 WMMA
// accumulator per wave. K consumed in chunks of 32 (8 x v_wmma_f32_16x16x4).
// Fast path: interior tiles staged with unconditional float4 loads.
// ---------------------------------------------------------------------------
#define MB 128
#define NBT 16
#define KC 32

template <bool AFFINE, bool RELU>
__global__ __launch_bounds__(256) void gemm_wmma_f32(
    const float* __restrict__ A, int lda,
    const float* __restrict__ W, int ldw,
    const float* __restrict__ bias,
    const float* __restrict__ in_scale, const float* __restrict__ in_shift,
    float* __restrict__ C, int ldc,
    int M, int K, int Nc)
{
  __shared__ float As[MB][KC + 1];   // +1 pad: conflict-free column reads
  __shared__ float Bs[KC][NBT];

  const int tid  = threadIdx.x;
  const int wave = tid >> 5;
  const int lane = tid & 31;
  const int half = lane >> 4;        // 0: lanes 0-15, 1: lanes 16-31
  const int mh   = lane & 15;
  const int row0 = blockIdx.x * MB;
  const int n0   = blockIdx.y * NBT;

  const bool fullM = (row0 + MB) <= M;
  const bool vec4  = ((lda & 3) == 0);

  // fast-path A staging coordinates: thread -> (row rbase + i*32, quad of 4 k)
  const int quad  = tid & 7;         // k-group: k = quad*4 .. quad*4+3
  const int rbase = tid >> 3;        // 0..31

  v8f acc = {};

  for (int k0 = 0; k0 < K; k0 += KC) {
    const bool fullK = (k0 + KC) <= K;

    if (fullM && fullK && vec4) {
      // ---- fast path: 4 x float4 per thread, no bounds checks
      #pragma unroll
      for (int i = 0; i < 4; ++i) {
        const int r = rbase + i * 32;
        const float* p = A + (long)(row0 + r) * lda + k0 + quad * 4;
        const float4 v = *(const float4*)p;
        float4 w = v;
        if (AFFINE) {
          const int gk = k0 + quad * 4;
          w.x = fmaf(v.x, in_scale[gk + 0], in_shift[gk + 0]);
          w.y = fmaf(v.y, in_scale[gk + 1], in_shift[gk + 1]);
          w.z = fmaf(v.z, in_scale[gk + 2], in_shift[gk + 2]);
          w.w = fmaf(v.w, in_scale[gk + 3], in_shift[gk + 3]);
        }
        As[r][quad * 4 + 0] = w.x;
        As[r][quad * 4 + 1] = w.y;
        As[r][quad * 4 + 2] = w.z;
        As[r][quad * 4 + 3] = w.w;
      }
      if (k0 + KC < K) {
        // hint the next K-chunk into cache (global_prefetch_b8)
        __builtin_prefetch(A + (long)(row0 + rbase) * lda + k0 + KC + quad * 4, 0, 0);
      }
    } else {
      // ---- ragged path: scalar guarded loads, zero padding
      #pragma unroll
      for (int i = 0; i < (MB * KC) / 256; ++i) {
        int idx = tid + i * 256;
        int r = idx >> 5;
        int k = idx & 31;
        int gr = row0 + r, gk = k0 + k;
        float v = 0.0f;
        if (gr < M && gk < K) {
          v = A[(long)gr * lda + gk];
          if (AFFINE) v = fmaf(v, in_scale[gk], in_shift[gk]);
        }
        As[r][k] = v;
      }
    }

    // ---- B tile (32x16): tiny, scalar coalesced loads, zero padded
    #pragma unroll
    for (int i = 0; i < (KC * NBT) / 256; ++i) {
      int idx = tid + i * 256;
      int k = idx >> 4;
      int n = idx & 15;
      int gk = k0 + k, gn = n0 + n;
      Bs[k][n] = (gk < K && gn < Nc) ? W[(long)gk * ldw + gn] : 0.0f;
    }
    __syncthreads();

    const int wr = wave * 16;
    #pragma unroll
    for (int kk = 0; kk < KC; kk += 4) {
      // A frag 16x4: lanes 0-15 hold (K=kk,kk+1), lanes 16-31 hold (K=kk+2,kk+3)
      v2f a, b;
      a.x = As[wr + mh][kk + 2 * half];
      a.y = As[wr + mh][kk + 2 * half + 1];
      // B frag 4x16: mirrored K striping across lane halves
      b.x = Bs[kk + 2 * half][mh];
      b.y = Bs[kk + 2 * half + 1][mh];
      acc = __builtin_amdgcn_wmma_f32_16x16x4_f32(
          false, a, false, b, (short)0, acc, false, false);
    }
    __syncthreads();
  }

  // ---- epilogue: C layout v[r]: lanes0-15 -> M=r, lanes16-31 -> M=r+8
  const int col = n0 + mh;
  if (col < Nc) {
    const float bv = bias ? bias[col] : 0.0f;
    #pragma unroll
    for (int r = 0; r < 8; ++r) {
      int row = row0 + wave * 16 + r + 8 * half;
      if (row < M) {
        float v = acc[r] + bv;
        if (RELU) v = fmaxf(v, 0.0f);
        C[(long)row * ldc + col] = v;
      }
    }
  }
}

// ---------------------------------------------------------------------------
// Per-head attention logits: el[n,h] = sum_d feat[n,h*16+d] * al[h,d]
// ---------------------------------------------------------------------------
__global__ void attn_proj_kernel(const float* __restrict__ feat,
                                 const float* __restrict__ al,
                                 const float* __restrict__ ar,
                                 float* __restrict__ el, float* __restrict__ er,
                                 int n)
{
  int t = blockIdx.x * blockDim.x + threadIdx.x;
  if (t >= n * 8) return;
  int node = t >> 3, h = t & 7;
  const float* f = feat + (long)node * 128 + h * 16;
  float a = 0.0f, b = 0.0f;
  #pragma unroll
  for (int d = 0; d < 16; ++d) {
    float v = f[d];
    a += v * al[h * 16 + d];
    b += v * ar[h * 16 + d];
  }
  el[t] = a;
  er[t] = b;
}

// ordered-int encoding of f32 (monotone, self-inverse) for atomicMax
__device__ __forceinline__ int fkey(float f) {
  int i = __float_as_int(f);
  return i ^ ((i >> 31) & 0x7fffffff);
}
__device__ __forceinline__ float fdec(int k) {
  int i = k ^ ((k >> 31) & 0x7fffffff);
  return __int_as_float(i);
}

__global__ void init_gat_kernel(int* __restrict__ mkey, float* __restrict__ s,
                                float* __restrict__ accum, int n)
{
  long t = (long)blockIdx.x * blockDim.x + threadIdx.x;
  if (t < (long)n * 8) { mkey[t] = (int)0x80000000; s[t] = 0.0f; }
  if (t < (long)n * 128) accum[t] = 0.0f;
}

// edge pass 1: per-(edge,head) segment-max of leaky_relu(el[src]+er[dst])
__global__ void edge_max_kernel(const int* __restrict__ src,
                                const int* __restrict__ dst,
                                const float* __restrict__ el,
                                const float* __restrict__ er,
                                int* __restrict__ mkey, long nwork)
{
  long t = (long)blockIdx.x * blockDim.x + threadIdx.x;
  if (t >= nwork) return;
  int e = (int)(t >> 3), h = (int)(t & 7);
  int si = src[e], di = dst[e];
  float x = el[si * 8 + h] + er[di * 8 + h];
  x = x >= 0.0f ? x : 0.2f * x;
  atomicMax(&mkey[di * 8 + h], fkey(x));
}

// edge pass 2 (fused): ex = exp(e - m[dst]); accum[dst] += ex*feat[src];
// s[dst,h] += ex.  One block per edge, 128 lanes = feature channels.
__global__ __launch_bounds__(128) void edge_scatter_kernel(
    const int* __restrict__ src, const int* __restrict__ dst,
    const float* __restrict__ el, const float* __restrict__ er,
    const int* __restrict__ mkey, const float* __restrict__ feat,
    float* __restrict__ accum, float* __restrict__ s)
{
  int e = blockIdx.x;
  int c = threadIdx.x;
  int h = c >> 4;
  int si = src[e], di = dst[e];
  float x = el[si * 8 + h] + er[di * 8 + h];
  x = x >= 0.0f ? x : 0.2f * x;
  float m = fdec(mkey[di * 8 + h]);
  float ex = expf(x - m);
  atomicAdd(&accum[(long)di * 128 + c], ex * feat[(long)si * 128 + c]);
  if ((c & 15) == 0) atomicAdd(&s[di * 8 + h], ex);
}

// normalize + bias: accum[n,c] = accum[n,c]/s[n,c/16] + b_gat[c]
__global__ void gat_fin_kernel(float* __restrict__ accum,
                               const float* __restrict__ s,
                               const float* __restrict__ bias, int n)
{
  long t = (long)blockIdx.x * blockDim.x + threadIdx.x;
  if (t >= (long)n * 128) return;
  int node = (int)(t >> 7), c = (int)(t & 127);
  float sv = s[node * 8 + (c >> 4)];
  float v = (sv > 0.0f) ? accum[t] / sv : 0.0f;
  accum[t] = v + bias[c];
}

// ---------------------------------------------------------------------------
// BatchNorm (training-mode, biased var): two-level reduction + finalize
// ---------------------------------------------------------------------------
__global__ void bn_zero_kernel(float* __restrict__ sums, int n2c)
{
  int t = blockIdx.x * blockDim.x + threadIdx.x;
  if (t < n2c) sums[t] = 0.0f;
}

__global__ __launch_bounds__(256) void bn_stats_kernel(
    const float* __restrict__ x, int ldx, int C, int n, int rpb,
    float* __restrict__ sums)  // sums[0..C) = sum, sums[C..2C) = sumsq
{
  int r0 = blockIdx.x * rpb;
  int r1 = min(n, r0 + rpb);
  for (int c = threadIdx.x; c < C; c += blockDim.x) {
    float sm = 0.0f, sq = 0.0f;
    for (int r = r0; r < r1; ++r) {
      float v = x[(long)r * ldx + c];
      sm += v;
      sq += v * v;
    }
    atomicAdd(&sums[c], sm);
    atomicAdd(&sums[C + c], sq);
  }
}

__global__ void bn_finalize_kernel(const float* __restrict__ sums,
                                   const float* __restrict__ g,
                                   const float* __restrict__ b,
                                   float* __restrict__ ascale,
                                   float* __restrict__ bshift,
                                   int C, float invN)
{
  int c = blockIdx.x * blockDim.x + threadIdx.x;
  if (c >= C) return;
  float mu  = sums[c] * invN;
  float var = sums[C + c] * invN - mu * mu;
  float a = g[c] * rsqrtf(var + 1e-5f);
  ascale[c] = a;
  bshift[c] = b[c] - mu * a;
}

// y[n,(ldy)c] = x[n,(ldx)c]*a[c] + bsh[c]   (writes layer output into JK slice)
__global__ void affine_kernel(const float* __restrict__ x, int ldx,
                              float* __restrict__ y, int ldy,
                              const float* __restrict__ a,
                              const float* __restrict__ bsh, int n, int C)
{
  long t = (long)blockIdx.x * blockDim.x + threadIdx.x;
  if (t >= (long)n * C) return;
  int r = (int)(t / C), c = (int)(t % C);
  y[(long)r * ldy + c] = x[(long)r * ldx + c] * a[c] + bsh[c];
}

// out[n] = sum_c relu(x[n,c]*a[c]+bsh[c]) * w2[c]
__global__ void dec_final_kernel(const float* __restrict__ x,
                                 const float* __restrict__ a,
                                 const float* __restrict__ bsh,
                                 const float* __restrict__ w2,
                                 float* __restrict__ out, int n)
{
  int t = blockIdx.x * blockDim.x + threadIdx.x;
  if (t >= n) return;
  const float* row = x + (long)t * 128;
  float acc = 0.0f;
  #pragma unroll 4
  for (int c = 0; c < 128; ++c) {
    float v = row[c] * a[c] + bsh[c];
    v = fmaxf(v, 0.0f);
    acc += v * w2[c];
  }
  out[t] = acc;
}

// ---------------------------------------------------------------------------
extern "C" void kernel_launch(void* const* d_in, const int* in_sizes, int n_in,
                              void* d_out, int out_size, void* d_ws, size_t ws_size,
                              hipStream_t stream)
{
  (void)in_sizes; (void)n_in; (void)out_size; (void)ws_size;

  const float* x       = (const float*)d_in[0];
  const int*   src     = (const int*)d_in[1];
  const int*   dst     = (const int*)d_in[2];
  const float* W_emb   = (const float*)d_in[3];
  const float* b_emb   = (const float*)d_in[4];
  const float* W_gat   = (const float*)d_in[5];
  const float* attn_l  = (const float*)d_in[6];
  const float* attn_r  = (const float*)d_in[7];
  const float* b_gat   = (const float*)d_in[8];
  const float* bn1_g   = (const float*)d_in[9];
  const float* bn1_b   = (const float*)d_in[10];
  const float* ff_W1   = (const float*)d_in[11];
  const float* ff_b1   = (const float*)d_in[12];
  const float* ff_W2   = (const float*)d_in[13];
  const float* ff_b2   = (const float*)d_in[14];
  const float* bn2_g   = (const float*)d_in[15];
  const float* bn2_b   = (const float*)d_in[16];
  const float* dec_W1  = (const float*)d_in[17];
  const float* dec_bn_g= (const float*)d_in[18];
  const float* dec_bn_b= (const float*)d_in[19];
  const float* dec_W2  = (const float*)d_in[20];
  float* out = (float*)d_out;

  const int N = GN, NE = GNE;
  const float invN = 1.0f / (float)N;
  const int HID_LD = 224;   // 218 padded to keep float4 fast path in ff2

  // workspace carve-up (floats)
  float* hcat  = (float*)d_ws;                 // [N,512] JK concat buffer
  float* feat  = hcat  + (size_t)N * 512;      // [N,128]
  float* accum = feat  + (size_t)N * 128;      // [N,128]
  float* hid   = accum + (size_t)N * 128;      // [N,224] (218 used)
  float* el    = hid   + (size_t)N * HID_LD;   // [N,8]
  float* er    = el    + (size_t)N * 8;        // [N,8]
  float* sden  = er    + (size_t)N * 8;        // [N,8] softmax denominators
  int*   mkey  = (int*)(sden + (size_t)N * 8); // [N,8] encoded segment max
  float* bnsums= (float*)(mkey + (size_t)N * 8); // [1024]
  float* bna   = bnsums + 1024;                // [512]
  float* bnb   = bna + 512;                    // [512]

  auto gemm = [&](const float* A, int lda, const float* W, int ldw,
                  const float* bias, const float* isc, const float* ish,
                  float* C, int ldc, int M, int K, int Nc, int relu) {
    dim3 g((M + MB - 1) / MB, (Nc + NBT - 1) / NBT);
    if (isc) {
      if (relu)
        hipLaunchKernelGGL((gemm_wmma_f32<true, true>), g, dim3(256), 0, stream,
                           A, lda, W, ldw, bias, isc, ish, C, ldc, M, K, Nc);
      else
        hipLaunchKernelGGL((gemm_wmma_f32<true, false>), g, dim3(256), 0, stream,
                           A, lda, W, ldw, bias, isc, ish, C, ldc, M, K, Nc);
    } else {
      if (relu)
        hipLaunchKernelGGL((gemm_wmma_f32<false, true>), g, dim3(256), 0, stream,
                           A, lda, W, ldw, bias, isc, ish, C, ldc, M, K, Nc);
      else
        hipLaunchKernelGGL((gemm_wmma_f32<false, false>), g, dim3(256), 0, stream,
                           A, lda, W, ldw, bias, isc, ish, C, ldc, M, K, Nc);
    }
  };

  const int gN128 = (N * 128 + 255) / 256;
  const int gN8   = (N * 8 + 255) / 256;
  const int bnBlocks = (N + 255) / 256;

  // 1) embedding -> JK slice 0
  gemm(x, 64, W_emb, 128, b_emb, nullptr, nullptr, hcat, 512, N, 64, 128, 0);

  for (int l = 0; l < 3; ++l) {
    const float* hin = hcat + (size_t)l * 128;  // lda = 512

    // GAT linear projection
    gemm(hin, 512, W_gat + (size_t)l * 128 * 128, 128, nullptr, nullptr, nullptr,
         feat, 128, N, 128, 128, 0);
    hipLaunchKernelGGL(attn_proj_kernel, dim3(gN8), dim3(256), 0, stream,
                       feat, attn_l + l * 128, attn_r + l * 128, el, er, N);
    hipLaunchKernelGGL(init_gat_kernel, dim3(gN128), dim3(256), 0, stream,
                       mkey, sden, accum, N);
    hipLaunchKernelGGL(edge_max_kernel,
                       dim3((unsigned)(((long)NE * 8 + 255) / 256)), dim3(256), 0, stream,
                       src, dst, el, er, mkey, (long)NE * 8);
    hipLaunchKernelGGL(edge_scatter_kernel, dim3(NE), dim3(128), 0, stream,
                       src, dst, el, er, mkey, feat, accum, sden);
    hipLaunchKernelGGL(gat_fin_kernel, dim3(gN128), dim3(256), 0, stream,
                       accum, sden, b_gat + l * 128, N);

    // BN1 (fused into FFN gemm1 A-load as per-channel affine)
    hipLaunchKernelGGL(bn_zero_kernel, dim3(1), dim3(256), 0, stream, bnsums, 256);
    hipLaunchKernelGGL(bn_stats_kernel, dim3(bnBlocks), dim3(256), 0, stream,
                       accum, 128, 128, N, 256, bnsums);
    hipLaunchKernelGGL(bn_finalize_kernel, dim3(1), dim3(128), 0, stream,
                       bnsums, bn1_g + l * 128, bn1_b + l * 128, bna, bnb, 128, invN);

    // FFN
    gemm(accum, 128, ff_W1 + (size_t)l * 128 * 218, 218, ff_b1 + l * 218,
         bna, bnb, hid, HID_LD, N, 128, 218, 1);
    gemm(hid, HID_LD, ff_W2 + (size_t)l * 218 * 128, 128, ff_b2 + l * 128,
         nullptr, nullptr, accum, 128, N, 218, 128, 0);

    // BN2 -> write into JK slice (l+1)
    hipLaunchKernelGGL(bn_zero_kernel, dim3(1), dim3(256), 0, stream, bnsums, 256);
    hipLaunchKernelGGL(bn_stats_kernel, dim3(bnBlocks), dim3(256), 0, stream,
                       accum, 128, 128, N, 256, bnsums);
    hipLaunchKernelGGL(bn_finalize_kernel, dim3(1), dim3(128), 0, stream,
                       bnsums, bn2_g + l * 128, bn2_b + l * 128, bna, bnb, 128, invN);
    hipLaunchKernelGGL(affine_kernel, dim3(gN128), dim3(256), 0, stream,
                       accum, 128, hcat + (size_t)(l + 1) * 128, 512, bna, bnb, N, 128);
  }

  // decoder: [N,512] @ [512,128]
  gemm(hcat, 512, dec_W1, 128, nullptr, nullptr, nullptr, feat, 128, N, 512, 128, 0);
  hipLaunchKernelGGL(bn_zero_kernel, dim3(1), dim3(256), 0, stream, bnsums, 256);
  hipLaunchKernelGGL(bn_stats_kernel, dim3(bnBlocks), dim3(256), 0, stream,
                     feat, 128, 128, N, 256, bnsums);
  hipLaunchKernelGGL(bn_finalize_kernel, dim3(1), dim3(128), 0, stream,
                     bnsums, dec_bn_g, dec_bn_b, bna, bnb, 128, invN);
  hipLaunchKernelGGL(dec_final_kernel, dim3((N + 127) / 128), dim3(128), 0, stream,
                     feat, bna, bnb, dec_W2, out, N);
}